// HarmonicPolynomialR3Generator_69191923138650
// MI455X (gfx1250) — compile-verified
//
#include <hip/hip_runtime.h>
#include <utility>

#define LMAX 10

typedef __attribute__((ext_vector_type(2))) float v2f;
typedef __attribute__((ext_vector_type(8))) float v8f;

__host__ __device__ constexpr int KofL(int l)      { return 3 * (2 * l - 1); }
__host__ __device__ constexpr int KstepsOfL(int l) { return (KofL(l) + 3) / 4; }
__host__ __device__ constexpr int MtilesOfL(int l) { return (2 * l + 1) > 16 ? 2 : 1; }
__host__ __device__ constexpr int TileBaseOfL(int l) {
    int t = 0;
    for (int j = 2; j < l; ++j) t += KstepsOfL(j) * MtilesOfL(j);
    return t;
}
constexpr int TOTAL_TILES = TileBaseOfL(LMAX + 1);   // 119 tiles
constexpr int NFEAT = (LMAX + 1) * (LMAX + 1);       // 121

template <int I, int N, class F>
__device__ inline void static_for(F&& f) {
    if constexpr (I < N) {
        f(std::integral_constant<int, I>{});
        static_for<I + 1, N>(static_cast<F&&>(f));
    }
}

struct HarmArgs {
    const float* points;
    const float* cob[LMAX];
    float* out;
    int N;
};

__global__ __launch_bounds__(256) void harmonic_wmma_kernel(HarmArgs args) {
    // A-tiles for v_wmma_f32_16x16x4_f32, pre-swizzled:
    // entry(lane, j): row = (lane&15) + 16*mtile, k = 4*kstep + 2*(lane>>4) + j
    __shared__ float ldsA[TOTAL_TILES * 64];

    for (int idx = threadIdx.x; idx < TOTAL_TILES * 32; idx += blockDim.x) {
        const int tile = idx >> 5;
        const int lane = idx & 31;
        int t = tile, l = 2, ks = 0, mt = 0;
        for (; l <= LMAX; ++l) {
            const int Kl = 3 * (2 * l - 1);
            const int kst = (Kl + 3) / 4;
            const int mtl = (2 * l + 1 > 16) ? 2 : 1;
            const int nt = kst * mtl;
            if (t < nt) { mt = t / kst; ks = t % kst; break; }
            t -= nt;
        }
        const int Kl   = 3 * (2 * l - 1);
        const int rows = 2 * l + 1;
        const int row  = (lane & 15) + mt * 16;
        const int kb   = ks * 4 + (lane >> 4) * 2;
        const float* cob = args.cob[l - 1];
        float v0 = 0.f, v1 = 0.f;
        if (row < rows) {
            if (kb < Kl)     v0 = cob[row * Kl + kb];
            if (kb + 1 < Kl) v1 = cob[row * Kl + kb + 1];
        }
        ldsA[tile * 64 + lane * 2 + 0] = v0;
        ldsA[tile * 64 + lane * 2 + 1] = v1;
    }
    __syncthreads();

    // cob_1 (3x3) is uniform -> scalar loads, kept in registers
    float c1[9];
#pragma unroll
    for (int i = 0; i < 9; ++i) c1[i] = args.cob[0][i];

    const int  lane = threadIdx.x & 31;
    const int  n    = lane & 15;          // point slot (both half-waves mirror)
    const bool hi   = lane >= 16;
    const int  wavesPerBlock = blockDim.x >> 5;
    const int  waveId     = blockIdx.x * wavesPerBlock + (threadIdx.x >> 5);
    const int  waveStride = gridDim.x * wavesPerBlock;
    const int  numGroups  = (args.N + 15) >> 4;

    for (int g = waveId; g < numGroups; g += waveStride) {
        const int  p     = g * 16 + n;
        const bool valid = p < args.N;

        float px = 0.f, py = 0.f, pz = 0.f;
        if (valid) {
            px = args.points[p * 3 + 0];
            py = args.points[p * 3 + 1];
            pz = args.points[p * 3 + 2];
        }
        float f1[3];
        f1[0] = c1[0] * px + c1[1] * py + c1[2] * pz;
        f1[1] = c1[3] * px + c1[4] * py + c1[5] * pz;
        f1[2] = c1[6] * px + c1[7] * py + c1[8] * pz;

        float* op = args.out + (size_t)p * NFEAT;
        if (valid && !hi) {
            op[0] = 1.0f;
            op[1] = f1[0];
            op[2] = f1[1];
            op[3] = f1[2];
        }

        float prev[2 * LMAX + 1];
        prev[0] = f1[0]; prev[1] = f1[1]; prev[2] = f1[2];

        static_for<0, LMAX - 1>([&](auto LC) {
            constexpr int LL   = decltype(LC)::value + 2;
            constexpr int K    = 3 * (2 * LL - 1);
            constexpr int KS   = (K + 3) / 4;
            constexpr int MT   = (2 * LL + 1 > 16) ? 2 : 1;
            constexpr int TB   = TileBaseOfL(LL);
            constexpr int ROWS = 2 * LL + 1;
            constexpr int OFF  = LL * LL;

            v8f acc[MT];
#pragma unroll
            for (int m = 0; m < MT; ++m)
                acc[m] = v8f{0.f, 0.f, 0.f, 0.f, 0.f, 0.f, 0.f, 0.f};

            static_for<0, KS>([&](auto KC) {
                constexpr int ks = decltype(KC)::value;
                constexpr int k0 = ks * 4;
                // B 4x16 tile: VGPR0 = K=k0 (lanes 0-15) | K=k0+2 (lanes 16-31)
                //              VGPR1 = K=k0+1            | K=k0+3
                float blo0, blo1, bhi0, bhi1;
                if constexpr (k0 + 0 < K) blo0 = prev[(k0 + 0) / 3] * f1[(k0 + 0) % 3]; else blo0 = 0.f;
                if constexpr (k0 + 1 < K) blo1 = prev[(k0 + 1) / 3] * f1[(k0 + 1) % 3]; else blo1 = 0.f;
                if constexpr (k0 + 2 < K) bhi0 = prev[(k0 + 2) / 3] * f1[(k0 + 2) % 3]; else bhi0 = 0.f;
                if constexpr (k0 + 3 < K) bhi1 = prev[(k0 + 3) / 3] * f1[(k0 + 3) % 3]; else bhi1 = 0.f;
                v2f b;
                b.x = hi ? bhi0 : blo0;
                b.y = hi ? bhi1 : blo1;

                static_for<0, MT>([&](auto MC) {
                    constexpr int mt = decltype(MC)::value;
                    const float* ap = &ldsA[(TB + mt * KS + ks) * 64 + lane * 2];
                    v2f a;
                    a.x = ap[0];
                    a.y = ap[1];
                    acc[mt] = __builtin_amdgcn_wmma_f32_16x16x4_f32(
                        false, a, false, b, (short)0, acc[mt], false, false);
                });
            });

            // Store features directly from D layout:
            // VGPR r holds M=r (lanes 0-15) and M=r+8 (lanes 16-31), N = lane%16
            static_for<0, MT>([&](auto MC) {
                constexpr int mt = decltype(MC)::value;
                static_for<0, 8>([&](auto RC) {
                    constexpr int r = decltype(RC)::value;
                    const int  row   = mt * 16 + r + (hi ? 8 : 0);
                    const bool rowOk = row < ROWS;
                    if (valid && rowOk) op[OFF + row] = acc[mt][r];
                });
            });

            // Re-gather D into per-lane prev[] (both half-waves) for next stage
            if constexpr (LL < LMAX) {
                static_for<0, MT>([&](auto MC) {
                    constexpr int mt = decltype(MC)::value;
                    static_for<0, 16>([&](auto RC) {
                        constexpr int r = decltype(RC)::value;
                        if constexpr (mt * 16 + r < ROWS) {
                            float v = __shfl(acc[mt][r & 7], (r < 8) ? n : (16 + n), 32);
                            prev[mt * 16 + r] = v;
                        }
                    });
                });
            }
        });
    }
}

extern "C" void kernel_launch(void* const* d_in, const int* in_sizes, int n_in,
                              void* d_out, int out_size, void* d_ws, size_t ws_size,
                              hipStream_t stream) {
    HarmArgs args;
    args.points = (const float*)d_in[0];
    for (int l = 1; l <= LMAX; ++l) args.cob[l - 1] = (const float*)d_in[l];
    args.out = (float*)d_out;
    args.N = in_sizes[0] / 3;

    const int block = 256;                     // 8 wave32 per block
    const int wavesPerBlock = block / 32;
    const int numGroups = (args.N + 15) / 16;
    int blocks = (numGroups + wavesPerBlock - 1) / wavesPerBlock;
    if (blocks > 2048) blocks = 2048;          // grid-stride; amortize LDS staging
    if (blocks < 1) blocks = 1;

    hipLaunchKernelGGL(harmonic_wmma_kernel, dim3(blocks), dim3(block), 0, stream, args);
}